// GRKAN_45586782879901
// MI455X (gfx1250) — compile-verified
//
#include <hip/hip_runtime.h>
#include <hip/hip_bf16.h>
#include <cstdint>

// ---------------------------------------------------------------------------
// GRKAN recurrence, restructured:
//   state = sum_s act(x_s@W+b) @ N_s,  N_s = M_s + diag(r) N_{s+1},
//   M_(e=T-1-s) = agg @ R^e = H_{e/32} @ P_{e%32},  P_i = R^i (log doubling).
// fp32 end-to-end via V_WMMA_F32_16X16X4_F32 (reference is fp32; memory-bound).
// ---------------------------------------------------------------------------

#define B_  256
#define T_  512
#define F_  256
#define NA_ 8
#define NO_ 128
#define TC_ 32      // timesteps per chunk in the main kernel
#define NCH (T_/TC_)

typedef __attribute__((ext_vector_type(2)))  float v2f;
typedef __attribute__((ext_vector_type(4)))  float v4f;
typedef __attribute__((ext_vector_type(8)))  float v8f;
typedef __attribute__((ext_vector_type(4)))  unsigned int v4u;
typedef __attribute__((ext_vector_type(8)))  int v8i;
typedef __attribute__((ext_vector_type(4)))  int v4i;

// ---- CDNA5 WMMA f32 16x16x4: D = A(16x4) * B(4x16) + C -------------------
__device__ __forceinline__ v8f wmma4(v2f a, v2f b, v8f c) {
    return __builtin_amdgcn_wmma_f32_16x16x4_f32(
        /*neg_a=*/false, a, /*neg_b=*/false, b,
        /*c_mod=*/(short)0, c, /*reuse_a=*/false, /*reuse_b=*/false);
}

// ---- Tensor Data Mover: 2D tile (rows x F_ f32, row stride T_*F_) -> LDS --
#if __has_builtin(__builtin_amdgcn_tensor_load_to_lds)
#define GRK_TDM 1
#else
#define GRK_TDM 0
#endif

#if GRK_TDM
__device__ __forceinline__ void tdm_load_tile(const float* gptr, void* ldsptr) {
    unsigned long long ga = (unsigned long long)(uintptr_t)gptr;
    unsigned la = (unsigned)(uintptr_t)ldsptr;     // low 32 bits = LDS offset
    v4u g0;
    g0.x = 1u;                                      // count=1, user mode, no gather
    g0.y = la;                                      // lds_addr [63:32]
    g0.z = (unsigned)(ga & 0xffffffffull);          // global_addr[31:0]
    g0.w = (unsigned)((ga >> 32) & 0x1ffffffull)    // global_addr[56:32]
         | (2u << 30);                              // type=2 ("image")
    v8i g1;
    g1[0] = (2 << 16);                              // data_size=4B; mask=0
    g1[1] = (int)((F_ & 0xffff) << 16);             // tensor_dim0 = 256 (lo16)
    g1[2] = (int)((16 & 0xffff) << 16);             // tensor_dim0 hi=0 | tensor_dim1=16
    g1[3] = (int)(F_ << 16);                        // tile_dim0 = 256
    g1[4] = 16;                                     // tile_dim1 = 16, tile_dim2 = 0
    g1[5] = T_ * F_;                                // tensor_dim0_stride (elements)
    g1[6] = 0;
    g1[7] = 0;
    v4i z4 = {0, 0, 0, 0};                          // groups 2/3 unused (2D tile)
    v8i z8 = {0, 0, 0, 0, 0, 0, 0, 0};              // extra group (clang-23 6-arg form)
    __builtin_amdgcn_tensor_load_to_lds(g0, g1, z4, z4, z8, 0);
}
#endif

__device__ __forceinline__ void plain_load_tile(const float* g, float* lds, int tid) {
    // 16 rows x 256 f32, row stride T_*F_ in global
    for (int i = tid; i < 16 * 64; i += 256) {
        int row = i >> 6, seg = i & 63;
        *(v4f*)(lds + row * 256 + seg * 4) =
            *(const v4f*)(g + (size_t)row * T_ * F_ + seg * 4);
    }
}

// ---- per-column activations (jax defaults; gelu approximate=tanh) ---------
__device__ __forceinline__ float activate(int j, float x) {
    switch (j) {
        case 0: case 6: return tanhf(x);
        case 1:         return fmaxf(x, 0.f);
        case 2: case 7: return 1.f / (1.f + expf(-x));
        case 3:         return x > 0.f ? x : expm1f(x);
        case 4: {       // gelu tanh-approx
            float u = 0.7978845608028654f * (x + 0.044715f * x * x * x);
            return 0.5f * x * (1.f + tanhf(u));
        }
        default:        return log1pf(expf(x));          // softplus
    }
}

// ===========================================================================
// K0: P_0 = I, P_1 = R
__global__ void __launch_bounds__(256) k_init_powers(const float* R, float* P) {
    int i = blockIdx.x * 256 + threadIdx.x;              // 16384 total
    int r = i >> 7, c = i & 127;
    P[i]         = (r == c) ? 1.f : 0.f;
    P[16384 + i] = R[i];
}

// K1: doubling round: block b computes P[m+1+b] = P[m] @ P[1+b]   (b<m)
__global__ void __launch_bounds__(256) k_power_round(float* P, int m) {
    const float* A  = P + (size_t)m * 16384;
    const float* Bm = P + (size_t)(1 + blockIdx.x) * 16384;
    float*       D  = P + (size_t)(m + 1 + blockIdx.x) * 16384;
    int tid = threadIdx.x, lane = tid & 31, w = tid >> 5;
    int lr = lane & 15, lh = lane >> 4;
    for (int tile = w; tile < 64; tile += 8) {
        int tm = tile >> 3, tn = tile & 7;
        v8f acc = {};
        for (int k = 0; k < 128; k += 4) {
            v2f a = *(const v2f*)(A + (tm * 16 + lr) * 128 + k + 2 * lh);
            v2f b;
            b.x = Bm[(k + 2 * lh)     * 128 + tn * 16 + lr];
            b.y = Bm[(k + 2 * lh + 1) * 128 + tn * 16 + lr];
            acc = wmma4(a, b, acc);
        }
        for (int v = 0; v < 8; ++v)
            D[(tm * 16 + v + 8 * lh) * 128 + tn * 16 + lr] = acc[v];
    }
}

// K2: H_k = agg @ (R^32)^k, k = 0..15   (tiny sequential chain, one block)
__global__ void __launch_bounds__(1024) k_hchain(const float* agg, const float* P,
                                                 float* H) {
    __shared__ float Hc[1024];
    const float* Rc = P + 32 * 16384;
    int tid = threadIdx.x;
    Hc[tid] = agg[tid];
    H[tid]  = agg[tid];
    __syncthreads();
    int j = tid >> 7, col = tid & 127;
    for (int k = 1; k < 16; ++k) {
        float s = 0.f;
        for (int q = 0; q < 128; ++q)
            s = fmaf(Hc[j * 128 + q], Rc[q * 128 + col], s);
        __syncthreads();
        Hc[tid] = s;
        H[k * 1024 + tid] = s;
        __syncthreads();
    }
}

// K3: M_e = H_{e/32} @ P_{e%32}   ([8,128]@[128,128], padded to 16 rows)
__global__ void __launch_bounds__(256) k_computeM(const float* H, const float* P,
                                                  float* M) {
    __shared__ float Hp[16 * 128];
    int e = blockIdx.x, tid = threadIdx.x;
    const float* Hk = H + (size_t)(e >> 5) * 1024;
    for (int i = tid; i < 2048; i += 256) Hp[i] = (i < 1024) ? Hk[i] : 0.f;
    __syncthreads();
    const float* Bm = P + (size_t)(e & 31) * 16384;
    int lane = tid & 31, w = tid >> 5, lr = lane & 15, lh = lane >> 4;
    v8f acc = {};
    for (int k = 0; k < 128; k += 4) {
        v2f a = *(const v2f*)(Hp + lr * 128 + k + 2 * lh);
        v2f b;
        b.x = Bm[(k + 2 * lh)     * 128 + w * 16 + lr];
        b.y = Bm[(k + 2 * lh + 1) * 128 + w * 16 + lr];
        acc = wmma4(a, b, acc);
    }
    if (lh == 0)                     // rows 8..15 are padding
        for (int v = 0; v < 8; ++v)
            M[(size_t)e * 1024 + v * 128 + w * 16 + lr] = acc[v];
}

// K4: backward scan N_s = M_{T-1-s} + r_j * N_{s+1}  (1024 independent chains)
__global__ void __launch_bounds__(1024) k_computeN(const float* M, const float* r,
                                                   float* N) {
    int tid = threadIdx.x;
    float rj = r[tid >> 7];
    float val = 0.f;
    for (int s = T_ - 1; s >= 0; --s) {
        int e = T_ - 1 - s;
        val = fmaf(rj, val, M[(size_t)e * 1024 + tid]);
        N[(size_t)s * 1024 + tid] = val;
    }
}

// K5: main streaming pass.  block = (btile, tchunk), 256 threads / 8 waves.
__global__ void __launch_bounds__(256) k_main(const float* X, const float* W,
                                              const float* bias, const float* Nm,
                                              float* Part) {
    __shared__ float Xb[2][16 * 256];   // double-buffered input tile (32 KB)
    __shared__ float Wl[256 * 16];      // W zero-padded to 16 cols (16 KB)
    __shared__ float pp[8 * 2 * 16 * 8];// split-K partials [w][lh][lr][v] (8 KB)
    __shared__ float dn[16 * 8];        // activated dense tile

    int tid = threadIdx.x, lane = tid & 31, w = tid >> 5;
    int lr = lane & 15, lh = lane >> 4;
    int btile = blockIdx.x, tch = blockIdx.y;
    int t0 = tch * TC_;
    const float* Xg = X + (size_t)btile * 16 * T_ * F_;   // row b0, t=0, f=0

    for (int i = tid; i < 4096; i += 256) {               // pad W: [256][16]
        int f = i >> 4, j = i & 15;
        Wl[i] = (j < 8) ? W[f * 8 + j] : 0.f;
    }

    // preload tile for t0
#if GRK_TDM
    if (w == 0) tdm_load_tile(Xg + (size_t)t0 * F_, &Xb[0][0]);
    if (w == 0) __builtin_amdgcn_s_wait_tensorcnt(0);
#else
    plain_load_tile(Xg + (size_t)t0 * F_, &Xb[0][0], tid);
#endif
    __syncthreads();

    v8f acc = {};                       // state tile: rows v+8*lh, cols w*16+lr
    int p = 0;
    for (int it = 0; it < TC_; ++it) {
        int t = t0 + it;
        if (it + 1 < TC_) {             // prefetch next tile into other buffer
#if GRK_TDM
            if (w == 0) tdm_load_tile(Xg + (size_t)(t + 1) * F_, &Xb[p ^ 1][0]);
#else
            plain_load_tile(Xg + (size_t)(t + 1) * F_, &Xb[p ^ 1][0], tid);
#endif
        }

        // ---- pre = Xtile @ Wpad, split-K over waves (wave w: K in [32w,32w+32))
        v8f pc = {};
        const float* Xr = &Xb[p][0];
        int k0 = w * 32;
        for (int kk = 0; kk < 32; kk += 4) {
            int k = k0 + kk;
            v2f a = *(const v2f*)(Xr + lr * 256 + k + 2 * lh);
            v2f b;
            b.x = Wl[(k + 2 * lh)     * 16 + lr];
            b.y = Wl[(k + 2 * lh + 1) * 16 + lr];
            pc = wmma4(a, b, pc);
        }
        {   // store C tile [row=v+8*lh, col=lr] as pp[w][lh][lr][v] (contiguous v)
            float* dst = &pp[((w * 2 + lh) * 16 + lr) * 8];
            *(v4f*)(dst)     = (v4f){pc[0], pc[1], pc[2], pc[3]};
            *(v4f*)(dst + 4) = (v4f){pc[4], pc[5], pc[6], pc[7]};
        }
        __syncthreads();

        // ---- reduce split-K partials + bias + per-column activation
        if (tid < 128) {
            int rr = tid >> 3, j = tid & 7;
            int vv = rr & 7, hh = rr >> 3;
            float s = bias[j];
            for (int ww = 0; ww < 8; ++ww)
                s += pp[((ww * 2 + hh) * 16 + j) * 8 + vv];
            dn[rr * 8 + j] = activate(j, s);
        }
        __syncthreads();

        // ---- state += dense(16x8) @ N_t(8x128); wave w owns cols [16w,16w+16)
        {
            const float* Nt = Nm + (size_t)t * 1024 + w * 16;
            for (int kk0 = 0; kk0 < 8; kk0 += 4) {
                v2f a = *(const v2f*)(dn + lr * 8 + kk0 + 2 * lh);
                v2f b;
                b.x = Nt[(kk0 + 2 * lh)     * 128 + lr];
                b.y = Nt[(kk0 + 2 * lh + 1) * 128 + lr];
                acc = wmma4(a, b, acc);
            }
        }
        __syncthreads();                       // all done reading Xb[p], dn
#if GRK_TDM
        if (it + 1 < TC_ && w == 0) __builtin_amdgcn_s_wait_tensorcnt(0);
#endif
        __syncthreads();                       // next tile visible to all waves
        p ^= 1;
    }

    // partial state for this t-chunk: Part[tch][256][128]
    float* Pd = Part + ((size_t)tch * B_ + btile * 16) * NO_ + w * 16;
    for (int v = 0; v < 8; ++v)
        Pd[(v + 8 * lh) * NO_ + lr] = acc[v];
}

// K6: deterministic cross-chunk reduction -> d_out [256][128] f32
__global__ void __launch_bounds__(256) k_reduce(const float* Part, float* out) {
    int i = blockIdx.x * 256 + threadIdx.x;    // 32768
    float s = 0.f;
    for (int c = 0; c < NCH; ++c) s += Part[(size_t)c * (B_ * NO_) + i];
    out[i] = s;
}

// ===========================================================================
extern "C" void kernel_launch(void* const* d_in, const int* in_sizes, int n_in,
                              void* d_out, int out_size, void* d_ws, size_t ws_size,
                              hipStream_t stream) {
    const float* X   = (const float*)d_in[0];  // [256,512,256]
    const float* W   = (const float*)d_in[1];  // [256,8]
    const float* b   = (const float*)d_in[2];  // [8]
    const float* r   = (const float*)d_in[3];  // [8]
    const float* agg = (const float*)d_in[4];  // [8,128]
    const float* R   = (const float*)d_in[5];  // [128,128]
    float* out = (float*)d_out;                // [256,128]

    float* ws = (float*)d_ws;
    float* P    = ws;                  // 33 * 16384   = 540672 f
    float* H    = P + 33 * 16384;      // 16 * 1024    = 16384 f
    float* M    = H + 16 * 1024;       // 512 * 1024   = 524288 f
    float* N    = M + 512 * 1024;      // 512 * 1024   = 524288 f
    float* Part = N + 512 * 1024;      // 16*256*128   = 524288 f   (~8.5 MB total)

    k_init_powers<<<64, 256, 0, stream>>>(R, P);
    for (int m = 1; m <= 16; m <<= 1)
        k_power_round<<<m, 256, 0, stream>>>(P, m);
    k_hchain<<<1, 1024, 0, stream>>>(agg, P, H);
    k_computeM<<<512, 256, 0, stream>>>(H, P, M);
    k_computeN<<<1, 1024, 0, stream>>>(M, r, N);
    k_main<<<dim3(B_ / 16, NCH), 256, 0, stream>>>(X, W, b, N, Part);
    k_reduce<<<(B_ * NO_) / 256, 256, 0, stream>>>(Part, out);
    (void)in_sizes; (void)n_in; (void)out_size; (void)ws_size;
}